// Generating_diffusion_model_61366492725508
// MI455X (gfx1250) — compile-verified
//
#include <hip/hip_runtime.h>

typedef __attribute__((ext_vector_type(8))) float v8f;
typedef __attribute__((ext_vector_type(2))) float v2f;

// D = A(16x4 f32) x B(4x16 f32) + C(16x16 f32), wave32
// 8-arg pattern: (neg_a, A, neg_b, B, c_mod, C, reuse_a, reuse_b)
#define WMMA_F32X4(a, b, c) \
  __builtin_amdgcn_wmma_f32_16x16x4_f32(false, (a), false, (b), (short)0, (c), false, false)

// Per layer: rho' = sum_k K_k rho K_k^dagger, complex f32, D=16, K=5.
// One wave per batch sample. All matrix math on v_wmma_f32_16x16x4_f32.
__global__ __launch_bounds__(256) void hqmm_kraus_kernel(
    const float* __restrict__ sr, const float* __restrict__ si,
    const float* __restrict__ kr, const float* __restrict__ ki,
    float* __restrict__ out, int nB, int L)
{
    // per-wave LDS: [0]=rho^T real, [1]=rho^T imag, [2]=T real (row-major), [3]=T imag
    __shared__ float lds[8][4][256];

    const int wave = threadIdx.x >> 5;
    const int lane = threadIdx.x & 31;
    const int lh   = lane >> 4;   // lane half: 0 or 1
    const int ll   = lane & 15;   // lane within half
    const int b    = blockIdx.x * 8 + wave;
    if (b >= nB) return;          // whole-wave exit only; active waves keep EXEC all-1s

    float* rtr = lds[wave][0];
    float* rti = lds[wave][1];
    float* ttr = lds[wave][2];
    float* tti = lds[wave][3];

    // --- initial state: global row-major -> LDS transposed (rho^T) ---
    #pragma unroll
    for (int r = 0; r < 8; ++r) {
        const int row = r + 8 * lh, col = ll;
        rtr[col * 16 + row] = sr[b * 256 + row * 16 + col];
        rti[col * 16 + row] = si[b * 256 + row * 16 + col];
    }

    v8f accr = {0.f,0.f,0.f,0.f,0.f,0.f,0.f,0.f};
    v8f acci = {0.f,0.f,0.f,0.f,0.f,0.f,0.f,0.f};

    for (int layer = 0; layer < L; ++layer) {
        // B-operand chunks of rho: A-pattern reads of rho^T (contiguous b64 per lane)
        v2f rbr[4], rbi[4];
        #pragma unroll
        for (int j = 0; j < 4; ++j) {
            const int idx = ll * 16 + 4 * j + 2 * lh;
            rbr[j] = *(const v2f*)&rtr[idx];
            rbi[j] = *(const v2f*)&rti[idx];
        }

        accr = v8f{0.f,0.f,0.f,0.f,0.f,0.f,0.f,0.f};
        acci = v8f{0.f,0.f,0.f,0.f,0.f,0.f,0.f,0.f};

        const float* krl = kr + (size_t)layer * 5 * 256;
        const float* kil = ki + (size_t)layer * 5 * 256;

        for (int k = 0; k < 5; ++k) {
            // A-pattern chunks of K_k; same registers reused as B-operand of K^H
            v2f Kr[4], Kj[4], KjN[4];
            #pragma unroll
            for (int j = 0; j < 4; ++j) {
                const int idx = k * 256 + ll * 16 + 4 * j + 2 * lh;
                Kr[j]  = *(const v2f*)&krl[idx];
                Kj[j]  = *(const v2f*)&kil[idx];
                KjN[j] = -Kj[j];
            }

            // T = K * rho   (Tr = Kr*Pr - Ki*Pi ; Ti = Kr*Pi + Ki*Pr)
            v8f tr = {0.f,0.f,0.f,0.f,0.f,0.f,0.f,0.f};
            v8f ti = {0.f,0.f,0.f,0.f,0.f,0.f,0.f,0.f};
            #pragma unroll
            for (int j = 0; j < 4; ++j) {
                tr = WMMA_F32X4(Kr[j],  rbr[j], tr);
                tr = WMMA_F32X4(KjN[j], rbi[j], tr);
                ti = WMMA_F32X4(Kr[j],  rbi[j], ti);
                ti = WMMA_F32X4(Kj[j],  rbr[j], ti);
            }

            // spill T (C layout) to LDS row-major, reload as A-operand chunks
            #pragma unroll
            for (int r = 0; r < 8; ++r) {
                const int row = r + 8 * lh, col = ll;
                ttr[row * 16 + col] = tr[r];
                tti[row * 16 + col] = ti[r];
            }
            v2f TAr[4], TAi[4];
            #pragma unroll
            for (int j = 0; j < 4; ++j) {
                const int idx = ll * 16 + 4 * j + 2 * lh;
                TAr[j] = *(const v2f*)&ttr[idx];
                TAi[j] = *(const v2f*)&tti[idx];
            }

            // acc += T * K^H  (Ur = Tr*Kr^T + Ti*Ki^T ; Ui = Ti*Kr^T - Tr*Ki^T)
            // B-operand of Kr^T / Ki^T is exactly the Kr / Kj register data.
            #pragma unroll
            for (int j = 0; j < 4; ++j) {
                accr = WMMA_F32X4(TAr[j], Kr[j],  accr);
                accr = WMMA_F32X4(TAi[j], Kj[j],  accr);
                acci = WMMA_F32X4(TAr[j], KjN[j], acci);
                acci = WMMA_F32X4(TAi[j], Kr[j],  acci);
            }
        }

        if (layer + 1 < L) {
            // next layer's rho^T from accumulator (C layout -> transposed store)
            #pragma unroll
            for (int r = 0; r < 8; ++r) {
                const int row = r + 8 * lh, col = ll;
                rtr[col * 16 + row] = accr[r];
                rti[col * 16 + row] = acci[r];
            }
        }
    }

    // output [B,D,D,2] interleaved real/imag
    #pragma unroll
    for (int r = 0; r < 8; ++r) {
        const int row = r + 8 * lh, col = ll;
        v2f o;
        o.x = accr[r];
        o.y = acci[r];
        *(v2f*)&out[(size_t)(b * 256 + row * 16 + col) * 2] = o;
    }
}

extern "C" void kernel_launch(void* const* d_in, const int* in_sizes, int n_in,
                              void* d_out, int out_size, void* d_ws, size_t ws_size,
                              hipStream_t stream) {
    const float* sr = (const float*)d_in[0];   // state_real [B,16,16]
    const float* si = (const float*)d_in[1];   // state_imag [B,16,16]
    const float* kr = (const float*)d_in[2];   // kraus_real [L,5,16,16]
    const float* ki = (const float*)d_in[3];   // kraus_imag [L,5,16,16]
    float* out = (float*)d_out;                // [B,16,16,2]

    const int nB = in_sizes[0] / 256;
    const int L  = in_sizes[2] / (5 * 256);
    const int blocks = (nB + 7) / 8;

    hipLaunchKernelGGL(hqmm_kraus_kernel, dim3(blocks), dim3(256), 0, stream,
                       sr, si, kr, ki, out, nB, L);
}